// GaussianUpsample_21251498180967
// MI455X (gfx1250) — compile-verified
//
#include <hip/hip_runtime.h>
#include <hip/hip_bf16.h>

#define BATCH   16
#define LSEQ    512
#define DMODEL  256
#define WSTRIDE 520   // padded LDS row stride (dwords) to stagger banks vs 512

typedef __attribute__((ext_vector_type(2))) float v2f;
typedef __attribute__((ext_vector_type(8))) float v8f;

// ---------------------------------------------------------------------------
// Kernel 1: alignment[b, t] = (t - start[seg(t)] + 1) for t < total, else 0
// One 512-thread block per batch row; scan + upper_bound in LDS.
// ---------------------------------------------------------------------------
__global__ void GaussianUpsample_align_kernel(const int* __restrict__ dur,
                                              float* __restrict__ alignment,
                                              int T) {
    __shared__ int s_cum[LSEQ];
    __shared__ int s_start[LSEQ];
    const int b   = blockIdx.x;
    const int tid = threadIdx.x;

    int d = dur[b * LSEQ + tid];
    s_cum[tid] = d;
    __syncthreads();
    for (int off = 1; off < LSEQ; off <<= 1) {
        int v   = s_cum[tid];
        int add = (tid >= off) ? s_cum[tid - off] : 0;
        __syncthreads();
        s_cum[tid] = v + add;
        __syncthreads();
    }
    s_start[tid] = s_cum[tid] - d;
    __syncthreads();
    const int total = s_cum[LSEQ - 1];

    for (int t = tid; t < T; t += LSEQ) {
        // searchsorted(cum, t, side='right'): first j with cum[j] > t
        int lo = 0, hi = LSEQ;
        while (lo < hi) {
            int mid = (lo + hi) >> 1;
            if (s_cum[mid] <= t) lo = mid + 1; else hi = mid;
        }
        int j = (lo < LSEQ) ? lo : (LSEQ - 1);
        alignment[b * T + t] = (t < total) ? (float)(t - s_start[j] + 1) : 0.0f;
    }
}

// ---------------------------------------------------------------------------
// Kernel 2: out[b, t0:t0+16, :] = normalize(w) @ enc[b]
// 512 threads (16 waves). Weights (16 x 512 f32) staged in LDS once,
// reused by all 16 waves (one 16-wide d-tile each) via v_wmma_f32_16x16x4_f32.
// ---------------------------------------------------------------------------
__global__ void GaussianUpsample_wmma_kernel(const float* __restrict__ enc,
                                             const int*   __restrict__ dur,
                                             const float* __restrict__ rng,
                                             float* __restrict__ out,
                                             int T) {
    __shared__ int   s_cum[LSEQ];
    __shared__ float s_w[16 * WSTRIDE];
    __shared__ float s_invS[16];

    const int b      = blockIdx.y;
    const int t0     = blockIdx.x * 16;
    const int tid    = threadIdx.x;   // l = tid for phases 0/1
    const int waveId = tid >> 5;      // 0..15
    const int lane   = tid & 31;

    // ---- phase 0: inclusive scan of durations for this batch row ----
    int d = dur[b * LSEQ + tid];
    s_cum[tid] = d;
    __syncthreads();
    for (int off = 1; off < LSEQ; off <<= 1) {
        int v   = s_cum[tid];
        int add = (tid >= off) ? s_cum[tid - off] : 0;
        __syncthreads();
        s_cum[tid] = v + add;
        __syncthreads();
    }
    const float c   = (float)s_cum[tid] - 0.5f * (float)d;   // gaussian center
    const float rr  = rng[b * LSEQ + tid];
    const float ir2 = 1.0f / (rr * rr);

    // ---- phase 1: unnormalized weights, 16 t-rows x 512 l, into LDS ----
    #pragma unroll
    for (int m = 0; m < 16; ++m) {
        float dt = (float)(t0 + m) - c;
        s_w[m * WSTRIDE + tid] = __expf(-dt * dt * ir2);
    }
    __syncthreads();

    // per-row sums: wave w reduces row w (512 elems -> 16/lane -> shfl_xor)
    {
        float s = 0.0f;
        const float* row = &s_w[waveId * WSTRIDE];
        #pragma unroll
        for (int i = 0; i < LSEQ / 32; ++i) s += row[lane + 32 * i];
        #pragma unroll
        for (int m = 16; m >= 1; m >>= 1) s += __shfl_xor(s, m, 32);
        if (lane == 0) s_invS[waveId] = 1.0f / s;
    }
    __syncthreads();
    #pragma unroll
    for (int m = 0; m < 16; ++m) s_w[m * WSTRIDE + tid] *= s_invS[m];
    __syncthreads();

    // ---- phase 2: C(16x16) += A(16x4 of w) x B(4x16 of enc), K = L = 512 ----
    // f32 A layout: lanes 0-15 hold M=lane (K = vgpr), lanes 16-31 M=lane-16 (K = vgpr+2)
    // f32 B layout: lane&15 = N, K = vgpr + 2*(lane>>4)   (mirrors C/D striping)
    const int m    = lane & 15;        // output row within tile (A: M index)
    const int n    = lane & 15;        // output col within tile (B/D: N index)
    const int half = lane >> 4;
    const int d0   = waveId * 16;
    const float* encb = enc + (size_t)b * LSEQ * DMODEL;

    v8f acc = {};
    #pragma unroll 8
    for (int k0 = 0; k0 < LSEQ; k0 += 4) {
        const int ka = k0 + 2 * half;
        v2f a;
        a.x = s_w[m * WSTRIDE + ka];              // contiguous pair -> ds_load_2addr_b64
        a.y = s_w[m * WSTRIDE + ka + 1];
        v2f bm;
        bm.x = encb[(size_t)ka       * DMODEL + d0 + n];
        bm.y = encb[(size_t)(ka + 1) * DMODEL + d0 + n];
        acc = __builtin_amdgcn_wmma_f32_16x16x4_f32(
            /*neg_a=*/false, a, /*neg_b=*/false, bm,
            /*c_mod=*/(short)0, acc, /*reuse_a=*/false, /*reuse_b=*/false);
    }

    // D layout: VGPR j, lane -> (M = j + 8*(lane>>4), N = lane&15)
    // Fast path: full tile (all 16 t-rows valid) -> 8 straight b32 stores off
    // one base with immediate offsets. Only the last t-tile takes the
    // predicated slow path.
    float* p = out + ((size_t)b * T + (size_t)(t0 + 8 * half)) * DMODEL + d0 + n;
    if (t0 + 16 <= T) {
        #pragma unroll
        for (int j = 0; j < 8; ++j) p[(size_t)j * DMODEL] = acc[j];
    } else {
        #pragma unroll
        for (int j = 0; j < 8; ++j) {
            if (t0 + 8 * half + j < T) p[(size_t)j * DMODEL] = acc[j];
        }
    }
}

// ---------------------------------------------------------------------------
extern "C" void kernel_launch(void* const* d_in, const int* in_sizes, int n_in,
                              void* d_out, int out_size, void* d_ws, size_t ws_size,
                              hipStream_t stream) {
    const float* enc = (const float*)d_in[0];   // (B, L, D) f32
    const int*   dur = (const int*)  d_in[1];   // (B, L)    i32
    const float* rng = (const float*)d_in[2];   // (B, L, 1) f32
    // d_in[3] is T on device; under graph capture we cannot sync-copy it,
    // so derive it from the output size: out_size = B*T*(D+1).
    const int T = out_size / (BATCH * (DMODEL + 1));

    float* alignment = (float*)d_out;                       // (B, T, 1)
    float* out       = (float*)d_out + (size_t)BATCH * T;   // (B, T, D)

    GaussianUpsample_align_kernel<<<BATCH, LSEQ, 0, stream>>>(dur, alignment, T);

    const int tTiles = (T + 15) / 16;
    GaussianUpsample_wmma_kernel<<<dim3(tTiles, BATCH), LSEQ, 0, stream>>>(
        enc, dur, rng, out, T);
}